// LSR_tensor_dot_84172769068060
// MI455X (gfx1250) — compile-verified
//
#include <hip/hip_runtime.h>

// ---------------------------------------------------------------------------
// LSR tensor dot:  out[b] = <x[b], T>,  T = sum_s core x1 U1_s x2 U2_s x3 U3_s
// B=512, I=J=K=64, R=8, SEP=4.
//
// Phase 1 (tiny, WMMA f32 16x16x4, exact f32 math):
//   W_s[i,bc]  = sum_a U1_s[i,a] * core[a,bc]              (4 GEMMs 64x8x64)
//   T[i,jk]    = sum_s sum_bc W_s[i,bc] * U2_s[j,b2]*U3_s[k,c]
// Phase 2 (memory-bound, 512 MB stream of x):
//   partials -> deterministic per-batch reduction -> out[512]
// ---------------------------------------------------------------------------

typedef __attribute__((ext_vector_type(2))) float v2f;
typedef __attribute__((ext_vector_type(8))) float v8f;

#define KTOT   (64 * 64 * 64)      // 262144
#define NCHUNK 16
#define CHUNK  (KTOT / NCHUNK)     // 16384
#define THREADS 256
#define PER_THREAD (CHUNK / THREADS) // 64 floats = 16 float4 per thread

// ----------------------------------------------------------------------------
// Kernel 1a: W_s(64x64) = U1_s(64x8) @ core(8x64), per s. One wave per 16x16
// tile: grid = 4 s * 16 tiles. K=8 -> two v_wmma_f32_16x16x4_f32 steps.
// ----------------------------------------------------------------------------
__global__ __launch_bounds__(32) void build_W_kernel(
    const float* __restrict__ core,      // [8][8][8] -> [8][64]
    const float* __restrict__ factors,   // [4][3][64][8]
    float* __restrict__ W)               // [4][64][64]
{
    const int blk  = blockIdx.x;     // 0..63
    const int s    = blk >> 4;
    const int ti   = (blk >> 2) & 3; // row tile
    const int tj   = blk & 3;        // col tile
    const int lane = threadIdx.x;    // 0..31
    const int n    = lane & 15;
    const bool lo  = lane < 16;
    const int klo  = lo ? 0 : 2;

    const float* U1 = factors + ((s * 3 + 0) * 64) * 8;  // [64][8]

    v8f c = {};
#pragma unroll
    for (int kk = 0; kk < 2; ++kk) {
        const int kb = kk * 4 + klo;
        v2f a, b;
        // A 16x4: lane (m = n) holds A[m, kb], A[m, kb+1]
        a.x = U1[(ti * 16 + n) * 8 + kb];
        a.y = U1[(ti * 16 + n) * 8 + kb + 1];
        // B 4x16: lane (col n) holds B[kb, n], B[kb+1, n]
        b.x = core[kb * 64 + tj * 16 + n];
        b.y = core[(kb + 1) * 64 + tj * 16 + n];
        c = __builtin_amdgcn_wmma_f32_16x16x4_f32(false, a, false, b,
                                                  (short)0, c, false, false);
    }
#pragma unroll
    for (int r = 0; r < 8; ++r) {
        const int row = ti * 16 + r + (lo ? 0 : 8);
        W[s * 4096 + row * 64 + tj * 16 + n] = c[r];
    }
}

// ----------------------------------------------------------------------------
// Kernel 1b: T(64 x 4096) = sum_s W_s(64x64) @ Q_s(64x4096),
//            Q_s[bc, jk] = U2_s[j, bc>>3] * U3_s[k, bc&7]  (formed in regs).
// One wave per 16x16 output tile: grid = 4 i-tiles * 256 jk-tiles = 1024.
// 4 s * 16 K-steps = 64 WMMAs per tile.
// ----------------------------------------------------------------------------
__global__ __launch_bounds__(32) void build_T_kernel(
    const float* __restrict__ factors,   // [4][3][64][8]
    const float* __restrict__ W,         // [4][64][64]
    float* __restrict__ T)               // [64][4096]
{
    const int blk  = blockIdx.x;     // 0..1023
    const int it   = blk >> 8;       // i tile 0..3
    const int jt   = blk & 255;      // jk tile 0..255
    const int lane = threadIdx.x;
    const int n    = lane & 15;
    const bool lo  = lane < 16;
    const int klo  = lo ? 0 : 2;

    const int jk = jt * 16 + n;      // output column for this lane
    const int j  = jk >> 6;
    const int k  = jk & 63;

    v8f c = {};
    for (int s = 0; s < 4; ++s) {
        const float* U2 = factors + ((s * 3 + 1) * 64) * 8;  // [64][8]
        const float* U3 = factors + ((s * 3 + 2) * 64) * 8;  // [64][8]
        float u2[8], u3[8];
#pragma unroll
        for (int a = 0; a < 8; ++a) {
            u2[a] = U2[j * 8 + a];
            u3[a] = U3[k * 8 + a];
        }
        const float* Wrow = W + s * 4096 + (it * 16 + n) * 64;
#pragma unroll
        for (int kk = 0; kk < 16; ++kk) {
            const int bc = kk * 4;   // compile-time under unroll
            // lo lanes carry K = bc, bc+1 ; hi lanes carry K = bc+2, bc+3
            const float b0 = lo ? (u2[(bc + 0) >> 3] * u3[(bc + 0) & 7])
                                : (u2[(bc + 2) >> 3] * u3[(bc + 2) & 7]);
            const float b1 = lo ? (u2[(bc + 1) >> 3] * u3[(bc + 1) & 7])
                                : (u2[(bc + 3) >> 3] * u3[(bc + 3) & 7]);
            const float2 av = *(const float2*)(Wrow + bc + klo);  // 8B aligned
            v2f a, b;
            a.x = av.x; a.y = av.y;
            b.x = b0;   b.y = b1;
            c = __builtin_amdgcn_wmma_f32_16x16x4_f32(false, a, false, b,
                                                      (short)0, c, false, false);
        }
    }
#pragma unroll
    for (int r = 0; r < 8; ++r) {
        const int i = it * 16 + r + (lo ? 0 : 8);
        T[i * 4096 + jk] = c[r];
    }
}

// ----------------------------------------------------------------------------
// Kernel 2: memory-bound stream.  Block (b, ch) reduces x[b, ch*CHUNK .. )
// against T (L2-resident).  Coalesced float4 loads, fixed-tree reduction.
// ----------------------------------------------------------------------------
__global__ __launch_bounds__(THREADS) void dot_chunks_kernel(
    const float* __restrict__ x,         // [512][262144]
    const float* __restrict__ T,         // [262144]
    float* __restrict__ partials)        // [512 * NCHUNK]
{
    const int blk = blockIdx.x;          // b * NCHUNK + ch
    const int b   = blk >> 4;
    const int ch  = blk & (NCHUNK - 1);
    const int tid = threadIdx.x;

    const float4* xp = (const float4*)(x + (size_t)b * KTOT + (size_t)ch * CHUNK);
    const float4* tp = (const float4*)(T + (size_t)ch * CHUNK);

    float a0 = 0.f, a1 = 0.f, a2 = 0.f, a3 = 0.f;
#pragma unroll
    for (int itr = 0; itr < PER_THREAD / 4; ++itr) {   // 16 iterations
        const int idx = itr * THREADS + tid;
        if (itr < PER_THREAD / 4 - 4) {
            // stream-ahead hint for x: global_prefetch_b8
            __builtin_prefetch((const void*)(xp + idx + 4 * THREADS), 0, 0);
        }
        const float4 xv = xp[idx];
        const float4 tv = tp[idx];
        a0 = fmaf(xv.x, tv.x, a0);
        a1 = fmaf(xv.y, tv.y, a1);
        a2 = fmaf(xv.z, tv.z, a2);
        a3 = fmaf(xv.w, tv.w, a3);
    }
    float acc = (a0 + a1) + (a2 + a3);

    __shared__ float sdata[THREADS];
    sdata[tid] = acc;
    __syncthreads();
#pragma unroll
    for (int sft = THREADS / 2; sft > 0; sft >>= 1) {
        if (tid < sft) sdata[tid] += sdata[tid + sft];
        __syncthreads();
    }
    if (tid == 0) partials[blk] = sdata[0];
}

// ----------------------------------------------------------------------------
// Kernel 3: deterministic final reduction (fixed order, no float atomics).
// ----------------------------------------------------------------------------
__global__ __launch_bounds__(256) void reduce_out_kernel(
    const float* __restrict__ partials,  // [512 * NCHUNK]
    float* __restrict__ out)             // [512]
{
    const int b = blockIdx.x * 256 + threadIdx.x;
    if (b < 512) {
        float s = 0.f;
#pragma unroll
        for (int i = 0; i < NCHUNK; ++i) s += partials[b * NCHUNK + i];
        out[b] = s;
    }
}

// ----------------------------------------------------------------------------
extern "C" void kernel_launch(void* const* d_in, const int* in_sizes, int n_in,
                              void* d_out, int out_size, void* d_ws, size_t ws_size,
                              hipStream_t stream) {
    const float* x       = (const float*)d_in[0];  // [512,64,64,64]
    const float* core    = (const float*)d_in[1];  // [8,8,8]
    const float* factors = (const float*)d_in[2];  // [4,3,64,8]
    float* out = (float*)d_out;                    // [512]

    float* ws       = (float*)d_ws;
    float* T        = ws;                          // 262144 floats (1 MB)
    float* W        = ws + KTOT;                   // 16384 floats
    float* partials = ws + KTOT + 4 * 64 * 64;     // 8192 floats

    build_W_kernel<<<64, 32, 0, stream>>>(core, factors, W);
    build_T_kernel<<<1024, 32, 0, stream>>>(factors, W, T);
    dot_chunks_kernel<<<512 * NCHUNK, THREADS, 0, stream>>>(x, T, partials);
    reduce_out_kernel<<<2, 256, 0, stream>>>(partials, out);
}